// DKNN_24137716204250
// MI455X (gfx1250) — compile-verified
//
#include <hip/hip_runtime.h>
#include <hip/hip_bf16.h>

typedef float v2f __attribute__((ext_vector_type(2)));
typedef float v8f __attribute__((ext_vector_type(8)));

#define M_Q   32
#define N_NB  1024
#define K_DIM 128
#define TOPK  16

// ---------------------------------------------------------------------------
// Kernel 1: scores[m][n] = 2*dot(q_m, n_n) - ||n_n||^2   (||q||^2 dropped:
// softmax over i is invariant to a per-(S,m) constant shift of s).
// One wave (32 threads) per 16x16 output tile; K=128 via 32x wmma f32 16x16x4.
//
// A (16x4 f32) lane layout (ISA 7.12.2): lane l holds row M=l&15,
//   VGPR0 = A[M][kb], VGPR1 = A[M][kb+1], kb = (l>>4)*2.
// B (4x16 f32) mirrored: lane l holds col N=l&15, same K pair.
// C/D (16x16 f32): VGPR v, lane l -> M = v + 8*(l>>4), N = l&15.
// ---------------------------------------------------------------------------
__global__ __launch_bounds__(32)
void dknn_scores_wmma(const float* __restrict__ Q,
                      const float* __restrict__ NB,
                      float* __restrict__ scores)
{
    const int lane  = threadIdx.x;            // 0..31
    const int col   = lane & 15;              // M for A-load, N for B-load
    const int kb    = (lane >> 4) << 1;       // 0 or 2
    const int mBase = blockIdx.y * 16;        // 0 or 16
    const int nBase = blockIdx.x * 16;        // 0..1008

    const float* __restrict__ qrow = Q  + (size_t)(mBase + col) * K_DIM;
    const float* __restrict__ nrow = NB + (size_t)(nBase + col) * K_DIM;

    v8f acc = {};
    float nn = 0.0f;                          // partial ||n_col||^2 (this lane's K slice)

#pragma unroll
    for (int kk = 0; kk < K_DIM; kk += 4) {
        const v2f a = *(const v2f*)(qrow + kk + kb);   // global_load_b64
        const v2f b = *(const v2f*)(nrow + kk + kb);
        nn += b.x * b.x + b.y * b.y;
        // 8 args: (neg_a, A, neg_b, B, c_mod, C, reuse_a, reuse_b)
        acc = __builtin_amdgcn_wmma_f32_16x16x4_f32(
            false, a, false, b, (short)0, acc, false, false);
    }

    // Lane l and lane l^16 together cover the full K range for column `col`.
    nn += __shfl_xor(nn, 16, 32);

    const int rowOff = (lane >> 4) * 8;
#pragma unroll
    for (int v = 0; v < 8; ++v) {
        const int m = mBase + rowOff + v;
        scores[(size_t)m * N_NB + nBase + col] = 2.0f * acc[v] - nn;
    }
}

// ---------------------------------------------------------------------------
// Kernel 2: per (S,m) block: s = score + gumbel; B_i = sum_j |s_i - s_j|;
// out_i = sum_{r=1..16} softmax_i(c_r * s_i - B_i), c_r = n+1-2r, tau = 1.
// 256 threads = 8 waves; each wave owns 2 rank rows, reduces max/sum with
// wave32 shfl_xor, accumulates via LDS float atomics (ds_add_f32).
// ---------------------------------------------------------------------------
__global__ __launch_bounds__(256)
void dknn_topk(const float* __restrict__ scores,
               const float* __restrict__ gumbel,
               float* __restrict__ out)
{
    __shared__ __align__(16) float s_sh[N_NB];
    __shared__ __align__(16) float b_sh[N_NB];
    __shared__ __align__(16) float acc [N_NB];

    const int t  = threadIdx.x;
    const int sm = blockIdx.x;                 // 0..63  (= S*32 + m)
    const int m  = sm & (M_Q - 1);

    const float* __restrict__ g  = gumbel + (size_t)sm * N_NB;
    const float* __restrict__ sc = scores + (size_t)m  * N_NB;

    for (int i = t; i < N_NB; i += 256) {
        s_sh[i] = sc[i] + g[i];
        acc[i]  = 0.0f;
    }
    __syncthreads();

    // ---- B_i: 4 rows per thread, float4 LDS broadcast over j ----
    const float s0 = s_sh[t], s1 = s_sh[t + 256], s2 = s_sh[t + 512], s3 = s_sh[t + 768];
    float b0 = 0.f, b1 = 0.f, b2 = 0.f, b3 = 0.f;
    for (int j = 0; j < N_NB; j += 4) {
        const float4 sj = *(const float4*)(s_sh + j);   // ds_load_b128, broadcast
        b0 += fabsf(s0 - sj.x) + fabsf(s0 - sj.y) + fabsf(s0 - sj.z) + fabsf(s0 - sj.w);
        b1 += fabsf(s1 - sj.x) + fabsf(s1 - sj.y) + fabsf(s1 - sj.z) + fabsf(s1 - sj.w);
        b2 += fabsf(s2 - sj.x) + fabsf(s2 - sj.y) + fabsf(s2 - sj.z) + fabsf(s2 - sj.w);
        b3 += fabsf(s3 - sj.x) + fabsf(s3 - sj.y) + fabsf(s3 - sj.z) + fabsf(s3 - sj.w);
    }
    b_sh[t] = b0; b_sh[t + 256] = b1; b_sh[t + 512] = b2; b_sh[t + 768] = b3;
    __syncthreads();

    // ---- 16 rank-row softmaxes: wave w handles rows r = 2w+1, 2w+2 ----
    const int wave = t >> 5;
    const int lane = t & 31;

    for (int rr = 0; rr < 2; ++rr) {
        const int   r = wave * 2 + rr + 1;                       // 1..16
        const float c = (float)(N_NB + 1 - 2 * r);               // scaling_r

        float e[32];
        float lmax = -__builtin_inff();
#pragma unroll
        for (int k = 0; k < 32; ++k) {
            const int i = lane + 32 * k;
            const float L = c * s_sh[i] - b_sh[i];               // tau == 1
            e[k] = L;
            lmax = fmaxf(lmax, L);
        }
#pragma unroll
        for (int off = 16; off > 0; off >>= 1)
            lmax = fmaxf(lmax, __shfl_xor(lmax, off, 32));

        float lsum = 0.0f;
#pragma unroll
        for (int k = 0; k < 32; ++k) {
            e[k] = __expf(e[k] - lmax);                          // v_exp_f32
            lsum += e[k];
        }
#pragma unroll
        for (int off = 16; off > 0; off >>= 1)
            lsum += __shfl_xor(lsum, off, 32);

        const float inv = 1.0f / lsum;
#pragma unroll
        for (int k = 0; k < 32; ++k)
            atomicAdd(&acc[lane + 32 * k], e[k] * inv);          // ds_add_f32
    }
    __syncthreads();

    float* __restrict__ o = out + (size_t)sm * N_NB;
    for (int i = t; i < N_NB; i += 256)
        o[i] = acc[i];
}

// ---------------------------------------------------------------------------
extern "C" void kernel_launch(void* const* d_in, const int* in_sizes, int n_in,
                              void* d_out, int out_size, void* d_ws, size_t ws_size,
                              hipStream_t stream)
{
    const float* query     = (const float*)d_in[0];   // [32,128]
    const float* neighbors = (const float*)d_in[1];   // [1024,128]
    const float* gumbel    = (const float*)d_in[2];   // [2,32,1024]
    float*       out       = (float*)d_out;           // [2,32,1024]
    float*       scores    = (float*)d_ws;            // [32,1024] = 128 KB scratch

    dim3 gridS(N_NB / 16, M_Q / 16);                  // 64 x 2 tiles
    dknn_scores_wmma<<<gridS, 32, 0, stream>>>(query, neighbors, scores);

    dknn_topk<<<2 * M_Q, 256, 0, stream>>>(scores, gumbel, out);
}